// LinkPrediction_17248588660763
// MI455X (gfx1250) — compile-verified
//
#include <hip/hip_runtime.h>

typedef __attribute__((ext_vector_type(2))) float v2f;
typedef __attribute__((ext_vector_type(8))) float v8f;

constexpr int H  = 128;
constexpr int NP = 2048;
constexpr int ND = 1024;
constexpr int PPSTR = 132;   // padded stride for epilogue tile (conflict-free b32 reads)

// ---------------------------------------------------------------------------
// Pre-pass: proj[r][k] = sum_h emb[r][h] * W[h][k]   (W is 128x128 row-major)
// One wave computes a 16-row x 128-col tile with V_WMMA_F32_16X16X4_F32.
// B (Wi slice) staged in LDS in WMMA-B-ready packed float2 layout, two 32KB passes.
// ---------------------------------------------------------------------------
__global__ __launch_bounds__(256) void proj_kernel(const float* __restrict__ emb,
                                                   const float* __restrict__ W,
                                                   float* __restrict__ out) {
  __shared__ float sB[8192];                 // 32KB: float2[4 kt][32 hs][32 lane]

  const int tid  = threadIdx.x;
  const int lane = tid & 31;
  const int wave = tid >> 5;
  const int half = lane >> 4;
  const int colM = lane & 15;
  const int r0   = blockIdx.x * 128 + wave * 16;

  v8f c[8];
#pragma unroll
  for (int kt = 0; kt < 8; ++kt)
#pragma unroll
    for (int e = 0; e < 8; ++e) c[kt][e] = 0.0f;

#pragma unroll
  for (int pass = 0; pass < 2; ++pass) {
    // stage B for k-tiles [4*pass, 4*pass+4)
    for (int i = tid; i < 4096; i += 256) {
      const int l   = i & 31;
      const int hs  = (i >> 5) & 31;
      const int ktl = i >> 10;
      const int cc  = (4 * pass + ktl) * 16 + (l & 15);
      const int hh  = 4 * hs + 2 * (l >> 4);
      float2 v;
      v.x = W[hh * H + cc];
      v.y = W[(hh + 1) * H + cc];
      ((float2*)sB)[i] = v;
    }
    __syncthreads();

    const float2* Bp = (const float2*)sB;
#pragma unroll 4
    for (int hs = 0; hs < 32; ++hs) {
      const int hbase = hs * 4 + 2 * half;
      const float2 av = *(const float2*)&emb[(r0 + colM) * H + hbase];
      v2f a; a.x = av.x; a.y = av.y;
#pragma unroll
      for (int ktl = 0; ktl < 4; ++ktl) {
        const float2 bv = Bp[(ktl * 32 + hs) * 32 + lane];
        v2f b; b.x = bv.x; b.y = bv.y;
        c[4 * pass + ktl] = __builtin_amdgcn_wmma_f32_16x16x4_f32(
            false, a, false, b, (short)0, c[4 * pass + ktl], false, false);
      }
    }
    __syncthreads();   // before next restage
  }

#pragma unroll
  for (int kt = 0; kt < 8; ++kt)
#pragma unroll
    for (int v = 0; v < 8; ++v) {
      const int r = v + 8 * half;
      out[(r0 + r) * H + kt * 16 + colM] = c[kt][v];
    }
}

// ---------------------------------------------------------------------------
// Main kernel: block = 16 patients x 8 drugs (one drug per wave32).
// Per wave: C(16 patients x 128 k) = (p .* d) @ Wi via 256 f32 WMMAs,
// accumulators initialized with d_proj[m,k]+b1[k]; epilogue adds p_proj,
// relu, dot with W2, 16-lane reduce, sigmoid, store.
// ---------------------------------------------------------------------------
__global__ __launch_bounds__(256) void interact_kernel(
    const float* __restrict__ pat, const float* __restrict__ drg,
    const float* __restrict__ W1,  const float* __restrict__ b1,
    const float* __restrict__ W2,  const float* __restrict__ b2,
    const float* __restrict__ pproj, const float* __restrict__ dproj,
    float* __restrict__ out) {
  __shared__ float sB[8192];          // 32KB: packed B (Wi) for current k-half
  __shared__ float sPa[2048];         // 8KB: patient tile in A-ready float2 layout
  __shared__ float sPP[16 * PPSTR];   // p_proj tile, padded
  __shared__ float sD[8 * H];         // drug embeddings (broadcast reads)
  __shared__ float sDPB[8 * H];       // d_proj + b1 (accumulator init)
  __shared__ float sW2[H];

  const int tid  = threadIdx.x;
  const int lane = tid & 31;
  const int wave = tid >> 5;
  const int half = lane >> 4;
  const int colM = lane & 15;
  const int n0   = blockIdx.x * 16;
  const int m0   = blockIdx.y * 8;
  const float* Wi = W1 + 2 * H * H;

  // ---- stage per-block tiles ----
  for (int i = tid; i < 1024; i += 256) {          // patient tile, A layout
    const int l  = i & 31;
    const int hs = i >> 5;
    const int hh = 4 * hs + 2 * (l >> 4);
    float2 v;
    v.x = pat[(n0 + (l & 15)) * H + hh];
    v.y = pat[(n0 + (l & 15)) * H + hh + 1];
    ((float2*)sPa)[i] = v;
  }
  for (int i = tid; i < 16 * H; i += 256) {        // p_proj tile (padded)
    const int r = i >> 7, k = i & 127;
    sPP[r * PPSTR + k] = pproj[(n0 + r) * H + k];
  }
  for (int i = tid; i < 8 * H; i += 256) {         // drugs + (d_proj + b1)
    const int j = i >> 7, k = i & 127;
    sD[i]   = drg[(m0 + j) * H + k];
    sDPB[i] = dproj[(m0 + j) * H + k] + b1[k];
  }
  if (tid < H) sW2[tid] = W2[tid];
  __syncthreads();

  // ---- init accumulators with d_proj[m,k] + b1[k] (column-broadcast) ----
  v8f c[8];
#pragma unroll
  for (int kt = 0; kt < 8; ++kt) {
    const float iv = sDPB[wave * H + kt * 16 + colM];
#pragma unroll
    for (int e = 0; e < 8; ++e) c[kt][e] = iv;
  }

  const float2* Ap = (const float2*)sPa;
  const float2* Bp = (const float2*)sB;

#pragma unroll
  for (int pass = 0; pass < 2; ++pass) {
    // stage Wi k-tiles [4*pass, 4*pass+4) in packed B layout
    for (int i = tid; i < 4096; i += 256) {
      const int l   = i & 31;
      const int hs  = (i >> 5) & 31;
      const int ktl = i >> 10;
      const int cc  = (4 * pass + ktl) * 16 + (l & 15);
      const int hh  = 4 * hs + 2 * (l >> 4);
      float2 v;
      v.x = Wi[hh * H + cc];
      v.y = Wi[(hh + 1) * H + cc];
      ((float2*)sB)[i] = v;
    }
    __syncthreads();

#pragma unroll 4
    for (int hs = 0; hs < 32; ++hs) {
      const int hbase = hs * 4 + 2 * half;
      const float2 pv = Ap[hs * 32 + lane];
      const float2 dv = *(const float2*)&sD[wave * H + hbase];
      v2f a; a.x = pv.x * dv.x; a.y = pv.y * dv.y;   // A = p .* d[m]
#pragma unroll
      for (int ktl = 0; ktl < 4; ++ktl) {
        const float2 bv = Bp[(ktl * 32 + hs) * 32 + lane];
        v2f b; b.x = bv.x; b.y = bv.y;
        c[4 * pass + ktl] = __builtin_amdgcn_wmma_f32_16x16x4_f32(
            false, a, false, b, (short)0, c[4 * pass + ktl], false, false);
      }
    }
    __syncthreads();   // before restaging sB
  }

  // ---- epilogue: + p_proj, relu, * W2[k], reduce over k, sigmoid ----
  float part[8];
#pragma unroll
  for (int v = 0; v < 8; ++v) part[v] = 0.0f;

#pragma unroll
  for (int kt = 0; kt < 8; ++kt) {
    const float w2v = sW2[kt * 16 + colM];
#pragma unroll
    for (int v = 0; v < 8; ++v) {
      const int r = v + 8 * half;
      float val = c[kt][v] + sPP[r * PPSTR + kt * 16 + colM];
      val = fmaxf(val, 0.0f);
      part[v] += val * w2v;
    }
  }

  const float bb = b2[0];
  const int m = m0 + wave;
#pragma unroll
  for (int v = 0; v < 8; ++v) {
    float s = part[v];
    s += __shfl_xor(s, 1, 32);
    s += __shfl_xor(s, 2, 32);
    s += __shfl_xor(s, 4, 32);
    s += __shfl_xor(s, 8, 32);        // full sum within each 16-lane half
    if (colM == 0) {
      const int r = v + 8 * half;
      const float x = s + bb;
      out[(n0 + r) * ND + m] = 1.0f / (1.0f + expf(-x));
    }
  }
}

// ---------------------------------------------------------------------------
extern "C" void kernel_launch(void* const* d_in, const int* in_sizes, int n_in,
                              void* d_out, int out_size, void* d_ws, size_t ws_size,
                              hipStream_t stream) {
  const float* pat = (const float*)d_in[0];
  const float* drg = (const float*)d_in[1];
  const float* W1  = (const float*)d_in[2];
  const float* b1  = (const float*)d_in[3];
  const float* W2  = (const float*)d_in[4];
  const float* b2  = (const float*)d_in[5];
  float* out   = (float*)d_out;
  float* pproj = (float*)d_ws;               // 2048*128 f32 = 1 MB
  float* dproj = pproj + NP * H;             //  1024*128 f32 = 0.5 MB

  proj_kernel<<<NP / 128, 256, 0, stream>>>(pat, W1, pproj);            // Wp
  proj_kernel<<<ND / 128, 256, 0, stream>>>(drg, W1 + H * H, dproj);    // Wd
  interact_kernel<<<dim3(NP / 16, ND / 8), 256, 0, stream>>>(
      pat, drg, W1, b1, W2, b2, pproj, dproj, out);
}